// GraphAttentionConstructor_7224134992061
// MI455X (gfx1250) — compile-verified
//
#include <hip/hip_runtime.h>
#include <hip/hip_bf16.h>
#include <math.h>

// Problem constants (from reference)
#define BATCH      32
#define NUM_NODES  1024
#define INPUT_DIM  256
#define ATTN_DIM   128
#define TOPK       64

typedef float v2f __attribute__((ext_vector_type(2)));
typedef float v8f __attribute__((ext_vector_type(8)));

// -------- workspace layout (floats) --------
// [0, QK)            : q  [BATCH*NUM_NODES, ATTN_DIM]
// [QK, 2*QK)         : k  [BATCH*NUM_NODES, ATTN_DIM]
// [2*QK, +ROWS)      : kth sortable keys (as uint bits)
// [2*QK+ROWS, +ROWS) : d^{-1/2} per row
#define QK_ELEMS  ((size_t)BATCH * NUM_NODES * ATTN_DIM)   // 4,194,304
#define ROWS      (BATCH * NUM_NODES)                      // 32,768

__device__ __forceinline__ unsigned sortkey(float f) {
    unsigned u = __float_as_uint(f);
    return (u & 0x80000000u) ? ~u : (u | 0x80000000u);
}

// ---------------------------------------------------------------------------
// Kernel 1: q/k projections. One wave per 16x16 output tile.
//   q = x @ Wq^T + Wq_b + static_emb   (and same for k)
// x: [BATCH*NUM_NODES, INPUT_DIM] row-major, W: [ATTN_DIM, INPUT_DIM] row-major
// ---------------------------------------------------------------------------
__global__ void __launch_bounds__(128)
proj_qk_kernel(const float* __restrict__ x,
               const float* __restrict__ Wq_w, const float* __restrict__ Wq_b,
               const float* __restrict__ Wk_w, const float* __restrict__ Wk_b,
               const float* __restrict__ static_emb,
               float* __restrict__ qout, float* __restrict__ kout)
{
    const int lane   = threadIdx.x & 31;
    const int waveId = (blockIdx.x * (blockDim.x >> 5)) + (threadIdx.x >> 5);
    // 2 (q/k) * 2048 mtiles * 8 dtiles = 32768 waves
    const int which = waveId >> 14;           // 0 = q, 1 = k
    const int rem   = waveId & 16383;
    const int mt    = rem >> 3;               // 0..2047
    const int dt    = rem & 7;                // 0..7

    const float* W    = which ? Wk_w : Wq_w;
    const float* bias = which ? Wk_b : Wq_b;
    float*       outp = which ? kout : qout;

    const int m0  = mt * 16;
    const int d0  = dt * 16;
    const int l15 = lane & 15;
    const int hs  = lane >> 4;                // half-wave select: K-pair 0 or 1

    const float* arow = x + (size_t)(m0 + l15) * INPUT_DIM + 2 * hs;
    const float* brow = W + (size_t)(d0 + l15) * INPUT_DIM + 2 * hs;

    v8f acc = {};
#pragma unroll
    for (int kk = 0; kk < INPUT_DIM; kk += 4) {
        v2f a = *(const v2f*)(arow + kk);     // A: q-side rows, K = kk + 2*hs .. +1
        v2f b = *(const v2f*)(brow + kk);     // B: W^T columns (contiguous in W)
        acc = __builtin_amdgcn_wmma_f32_16x16x4_f32(
                  false, a, false, b, (short)0, acc, false, false);
    }

    // C/D layout: VGPR r -> M = m0 + r + 8*hs, N = d0 + (lane&15)
    const int d    = d0 + l15;
    const float bv = bias[d];
#pragma unroll
    for (int r = 0; r < 8; ++r) {
        const int m    = m0 + r + 8 * hs;
        const int node = m & (NUM_NODES - 1);
        outp[(size_t)m * ATTN_DIM + d] =
            acc[r] + bv + static_emb[(size_t)node * ATTN_DIM + d];
    }
}

// ---------------------------------------------------------------------------
// Kernel 2: scores = gelu((q . k^T) / sqrt(128)). One wave per 16x64 strip.
// Writes pre-topk adjacency into d_out.
// ---------------------------------------------------------------------------
__global__ void __launch_bounds__(128)
scores_kernel(const float* __restrict__ q, const float* __restrict__ k,
              float* __restrict__ out)
{
    const int lane   = threadIdx.x & 31;
    const int waveId = (blockIdx.x * (blockDim.x >> 5)) + (threadIdx.x >> 5);
    // BATCH * 64 mtiles * 16 ngroups = 32768 waves
    const int b   = waveId >> 10;
    const int rem = waveId & 1023;
    const int mt  = rem >> 4;                 // 0..63
    const int ng  = rem & 15;                 // 0..15 (64-col groups)

    const int m0 = mt * 16;
    const int n0 = ng * 64;
    const int l15 = lane & 15;
    const int hs  = lane >> 4;

    const float* qb = q + (size_t)b * NUM_NODES * ATTN_DIM;
    const float* kb = k + (size_t)b * NUM_NODES * ATTN_DIM;

    const float* arow = qb + (size_t)(m0 + l15) * ATTN_DIM + 2 * hs;
    const float* brow0 = kb + (size_t)(n0 + l15)      * ATTN_DIM + 2 * hs;
    const float* brow1 = kb + (size_t)(n0 + 16 + l15) * ATTN_DIM + 2 * hs;
    const float* brow2 = kb + (size_t)(n0 + 32 + l15) * ATTN_DIM + 2 * hs;
    const float* brow3 = kb + (size_t)(n0 + 48 + l15) * ATTN_DIM + 2 * hs;

    v8f acc0 = {}, acc1 = {}, acc2 = {}, acc3 = {};
#pragma unroll
    for (int kk = 0; kk < ATTN_DIM; kk += 4) {
        v2f a  = *(const v2f*)(arow + kk);
        v2f b0 = *(const v2f*)(brow0 + kk);
        v2f b1 = *(const v2f*)(brow1 + kk);
        v2f b2 = *(const v2f*)(brow2 + kk);
        v2f b3 = *(const v2f*)(brow3 + kk);
        acc0 = __builtin_amdgcn_wmma_f32_16x16x4_f32(false, a, false, b0, (short)0, acc0, false, false);
        acc1 = __builtin_amdgcn_wmma_f32_16x16x4_f32(false, a, false, b1, (short)0, acc1, false, false);
        acc2 = __builtin_amdgcn_wmma_f32_16x16x4_f32(false, a, false, b2, (short)0, acc2, false, false);
        acc3 = __builtin_amdgcn_wmma_f32_16x16x4_f32(false, a, false, b3, (short)0, acc3, false, false);
    }

    const float scale = 0.08838834764831845f;   // 1/sqrt(128)
    float* orow = out + ((size_t)b * NUM_NODES) * NUM_NODES;

#pragma unroll
    for (int r = 0; r < 8; ++r) {
        const int m = m0 + r + 8 * hs;
        float* op = orow + (size_t)m * NUM_NODES + n0 + l15;
        float s0 = acc0[r] * scale;
        float s1 = acc1[r] * scale;
        float s2 = acc2[r] * scale;
        float s3 = acc3[r] * scale;
        // exact (erf) GELU
        op[0]  = 0.5f * s0 * (1.0f + erff(s0 * 0.70710678118654752f));
        op[16] = 0.5f * s1 * (1.0f + erff(s1 * 0.70710678118654752f));
        op[32] = 0.5f * s2 * (1.0f + erff(s2 * 0.70710678118654752f));
        op[48] = 0.5f * s3 * (1.0f + erff(s3 * 0.70710678118654752f));
    }
}

// ---------------------------------------------------------------------------
// Kernel 3: per-row exact k-th largest (radix select on sortable bits) and
// degree d = sum(masked row) + 1. One 256-thread block per row.
// ---------------------------------------------------------------------------
__global__ void __launch_bounds__(256)
topk_degree_kernel(const float* __restrict__ adj,
                   unsigned* __restrict__ kth_key,
                   float* __restrict__ dinv)
{
    __shared__ unsigned keys[NUM_NODES];
    __shared__ float    vals[NUM_NODES];
    __shared__ int      cnt;
    __shared__ float    fsum;

    const int row = blockIdx.x;               // b*NUM_NODES + n
    const int tid = threadIdx.x;
    const float* rp = adj + (size_t)row * NUM_NODES;

    for (int i = tid; i < NUM_NODES; i += 256) {
        float v = rp[i];
        vals[i] = v;
        keys[i] = sortkey(v);
    }
    __syncthreads();

    // MSB-first radix select: max threshold t with count(keys >= t) >= TOPK
    unsigned prefix = 0u;
    for (int bit = 31; bit >= 0; --bit) {
        const unsigned cand = prefix | (1u << bit);
        if (tid == 0) cnt = 0;
        __syncthreads();
        int c = 0;
        for (int i = tid; i < NUM_NODES; i += 256) c += (keys[i] >= cand) ? 1 : 0;
        atomicAdd(&cnt, c);
        __syncthreads();
        if (cnt >= TOPK) prefix = cand;
        __syncthreads();
    }

    if (tid == 0) fsum = 0.0f;
    __syncthreads();
    float s = 0.0f;
    for (int i = tid; i < NUM_NODES; i += 256)
        if (keys[i] >= prefix) s += vals[i];
    atomicAdd(&fsum, s);
    __syncthreads();

    if (tid == 0) {
        kth_key[row] = prefix;
        dinv[row] = rsqrtf(fsum + 1.0f);      // +1 from self loop
    }
}

// ---------------------------------------------------------------------------
// Kernel 4: mask, add self loop, symmetric normalize. float4-vectorized.
// ---------------------------------------------------------------------------
__global__ void __launch_bounds__(256)
normalize_kernel(float* __restrict__ adj,
                 const unsigned* __restrict__ kth_key,
                 const float* __restrict__ dinv)
{
    const size_t t    = (size_t)blockIdx.x * blockDim.x + threadIdx.x;
    const size_t base = t * 4;                // 4 consecutive columns
    const int row = (int)(base >> 10);        // b*NUM_NODES + n
    const int m0  = (int)(base & (NUM_NODES - 1));
    const int b   = row >> 10;
    const int n   = row & (NUM_NODES - 1);

    const unsigned kk = kth_key[row];
    const float dr    = dinv[row];
    const float* dcol = dinv + (size_t)b * NUM_NODES;

    float4 v = *(float4*)(adj + base);
    float a[4] = { v.x, v.y, v.z, v.w };
#pragma unroll
    for (int j = 0; j < 4; ++j) {
        const int m = m0 + j;
        float x = (sortkey(a[j]) >= kk) ? a[j] : 0.0f;
        if (m == n) x += 1.0f;
        a[j] = x * dr * dcol[m];
    }
    v.x = a[0]; v.y = a[1]; v.z = a[2]; v.w = a[3];
    *(float4*)(adj + base) = v;
}

// ---------------------------------------------------------------------------
extern "C" void kernel_launch(void* const* d_in, const int* in_sizes, int n_in,
                              void* d_out, int out_size, void* d_ws, size_t ws_size,
                              hipStream_t stream)
{
    const float* x          = (const float*)d_in[0];
    const float* Wq_w       = (const float*)d_in[1];
    const float* Wq_b       = (const float*)d_in[2];
    const float* Wk_w       = (const float*)d_in[3];
    const float* Wk_b       = (const float*)d_in[4];
    const float* static_emb = (const float*)d_in[5];
    float* out = (float*)d_out;

    float*    ws_f    = (float*)d_ws;
    float*    qbuf    = ws_f;
    float*    kbuf    = ws_f + QK_ELEMS;
    unsigned* kth_key = (unsigned*)(ws_f + 2 * QK_ELEMS);
    float*    dinv    = ws_f + 2 * QK_ELEMS + ROWS;

    // 1) projections: 32768 waves, 4 waves/block
    proj_qk_kernel<<<8192, 128, 0, stream>>>(x, Wq_w, Wq_b, Wk_w, Wk_b,
                                             static_emb, qbuf, kbuf);
    // 2) scores + GELU: 32768 waves (16x64 strips), 4 waves/block
    scores_kernel<<<8192, 128, 0, stream>>>(qbuf, kbuf, out);
    // 3) per-row top-k threshold + degree
    topk_degree_kernel<<<ROWS, 256, 0, stream>>>(out, kth_key, dinv);
    // 4) mask + self loops + D^{-1/2} A D^{-1/2}
    normalize_kernel<<<(BATCH * NUM_NODES * NUM_NODES / 4) / 256, 256, 0, stream>>>(
        out, kth_key, dinv);
}